// DynEdgeConvGCNSegmentation_38019050504966
// MI455X (gfx1250) — compile-verified
//
#include <hip/hip_runtime.h>
#include <float.h>

#define NN    8192
#define NE    131072
#define KNEI  20
#define INDIM 3
#define HIDC  256
#define NCLS  32

typedef __attribute__((ext_vector_type(2))) float v2f;
typedef __attribute__((ext_vector_type(8))) float v8f;

__device__ __forceinline__ v8f wmma4(v2f a, v2f b, v8f c) {
  // V_WMMA_F32_16X16X4_F32 : D(16x16 f32) = A(16x4) * B(4x16) + C
  return __builtin_amdgcn_wmma_f32_16x16x4_f32(false, a, false, b, (short)0, c,
                                               false, false);
}

// gfx1250 async global->LDS copy (ASYNCcnt-tracked), b32 per lane.
__device__ __forceinline__ void async_copy_b32(unsigned ldsOff, const float* g) {
  asm volatile("global_load_async_to_lds_b32 %0, %1, off"
               :: "v"(ldsOff), "v"(g) : "memory");
}
__device__ __forceinline__ void wait_async() {
  asm volatile("s_wait_asynccnt 0" ::: "memory");
}

// ---------------- utility kernels ----------------
__global__ void k_zero(float* __restrict__ p, int n) {
  int i = blockIdx.x * blockDim.x + threadIdx.x;
  if (i < n) p[i] = 0.f;
}

__global__ void k_deg_accum(const int* __restrict__ src, const int* __restrict__ dst,
                            float* __restrict__ co, float* __restrict__ ci, int e) {
  int i = blockIdx.x * blockDim.x + threadIdx.x;
  if (i < e) {
    atomicAdd(&co[src[i]], 1.f);
    atomicAdd(&ci[dst[i]], 1.f);
  }
}

__global__ void k_deg_final(const float* __restrict__ cnt, float* __restrict__ d, int n) {
  int i = blockIdx.x * blockDim.x + threadIdx.x;
  if (i < n) d[i] = rsqrtf(fmaxf(cnt[i], 1.f));
}

// agg[dst[e]][c] += x[src[e]][c] * dout[src[e]]
__global__ void k_gconv_scatter(const float* __restrict__ X, const int* __restrict__ src,
                                const int* __restrict__ dst, const float* __restrict__ dO,
                                float* __restrict__ agg, int C, long total) {
  long t = (long)blockIdx.x * blockDim.x + threadIdx.x;
  if (t >= total) return;
  int e = (int)(t / C), c = (int)(t % C);
  int s = src[e], d = dst[e];
  atomicAdd(&agg[(long)d * C + c], X[(long)s * C + c] * dO[s]);
}

__global__ void k_sqnorm(const float* __restrict__ X, float* __restrict__ sq,
                         int n, int H) {
  int i = blockIdx.x * blockDim.x + threadIdx.x;
  if (i >= n) return;
  float s = 0.f;
  const float* row = X + (long)i * H;
  for (int c = 0; c < H; ++c) { float v = row[c]; s += v * v; }
  sq[i] = s;
}

// ---------------- WMMA GEMM ----------------
// Block = 128 threads = 4 waves stacked along M (block tile 64M x 64N).
// Each wave: 1 M-tile x 4 N-tiles (4 v8f accumulators), A fragment reused x4.
// B K-panel (64K x 64N) staged in LDS (transposed [n][k], stride 66 -> no bank
// conflicts) via async global->LDS; all 4 waves share it.
//
// AMODE 0: A[r][k] plain                            (Gram)
// AMODE 1: A[r][k] * rowScale[r]                    (GCN normalization)
// AMODE 2: A[r][k] = k<H ? X[i][k] : X[j][k-H]-X[i][k-H], i=r/K, j=nbr[r]
// AMODE 3: A[r][k] = relu(A[r][k]*bnS[k]+bnO[k])    (BN+ReLU of prev layer)
// GRAM: B indexed row-major [n][k] (ldb), epilogue sq_i+sq_j-2*acc into Cout.
template <int AMODE, bool RELU, bool GRAM>
__global__ void k_wmma_gemm(const float* __restrict__ A, const float* __restrict__ B,
                            const float* __restrict__ bias, float* __restrict__ Cout,
                            int M, int Kd, int Nc, int ldb,
                            const float* __restrict__ rowScale,
                            const float* __restrict__ X, const int* __restrict__ nbr,
                            int H,
                            const float* __restrict__ bnS, const float* __restrict__ bnO) {
  constexpr int KP  = 64;   // K panel depth
  constexpr int WN  = 64;   // N strip width
  constexpr int STR = 66;   // padded LDS row stride (floats)
  __shared__ float Bsm[WN * STR];

  const int tid  = threadIdx.x;
  const int lane = tid & 31;
  const int wave = tid >> 5;
  const int lm   = lane & 15;
  const int half = lane >> 4;
  const int row0 = (blockIdx.x * 4 + wave) * 16;
  const int colBase = blockIdx.y * WN;
  const int r = row0 + lm;  // A-row this lane feeds

  long ei = 0, ej = 0;
  float rsc = 1.f;
  if constexpr (AMODE == 2) { ei = (long)(r / KNEI) * H; ej = (long)nbr[r] * H; }
  if constexpr (AMODE == 1) { rsc = rowScale[r]; }

  auto fetchA = [&](int k) -> float {
    if (k >= Kd) return 0.f;
    if constexpr (AMODE == 0) {
      return A[(long)r * Kd + k];
    } else if constexpr (AMODE == 1) {
      return A[(long)r * Kd + k] * rsc;
    } else if constexpr (AMODE == 2) {
      if (k < H) return X[ei + k];
      int kk = k - H;
      return X[ej + kk] - X[ei + kk];
    } else {
      float v = A[(long)r * Kd + k] * bnS[k] + bnO[k];
      return v > 0.f ? v : 0.f;
    }
  };

  v8f acc[4] = {{}, {}, {}, {}};

  for (int kp = 0; kp < Kd; kp += KP) {
    // ---- stage B panel into LDS (transposed [n][k]) ----
    for (int e = tid; e < WN * KP; e += 128) {
      int k, n;
      if constexpr (GRAM) { k = e & 63; n = e >> 6; }   // coalesce along k
      else                { n = e & 63; k = e >> 6; }   // coalesce along n
      const int gk = kp + k, gn = colBase + n;
      const int off = n * STR + k;
      if (gk < Kd && gn < Nc) {
        const float* src = GRAM ? &B[(long)gn * ldb + gk]
                                : &B[(long)gk * Nc + gn];
        async_copy_b32((unsigned)(uintptr_t)&Bsm[off], src);
      } else {
        Bsm[off] = 0.f;
      }
    }
    wait_async();
    __syncthreads();

    // ---- compute over panel ----
    int kend = kp + KP;
    if (kend > Kd) kend = (Kd + 3) & ~3;  // ragged Kd handled by zero fill
    for (int k0 = kp; k0 < kend; k0 += 4) {
      const int ka = (k0 - kp) + 2 * half;  // local k in panel (even)
      v2f av;
      av.x = fetchA(k0 + 2 * half);
      av.y = fetchA(k0 + 2 * half + 1);
#pragma unroll
      for (int t = 0; t < 4; ++t) {
        const int nloc = t * 16 + lm;
        v2f bv;
        bv.x = Bsm[nloc * STR + ka];
        bv.y = Bsm[nloc * STR + ka + 1];
        acc[t] = wmma4(av, bv, acc[t]);
      }
    }
    __syncthreads();
  }

  // ---- epilogue ----
  const int orow = row0 + half * 8;
#pragma unroll
  for (int t = 0; t < 4; ++t) {
    if (colBase + t * 16 >= Nc) continue;  // uniform per wave
    const int cN = colBase + t * 16 + lm;
    if constexpr (GRAM) {
      const float sj = bias[cN];  // bias == sq (squared norms)
#pragma unroll
      for (int rr = 0; rr < 8; ++rr)
        Cout[(long)(orow + rr) * Nc + cN] = bias[orow + rr] + sj - 2.f * acc[t][rr];
    } else {
      const float bb = bias[cN];
#pragma unroll
      for (int rr = 0; rr < 8; ++rr) {
        float v = acc[t][rr] + bb;
        if (RELU) v = fmaxf(v, 0.f);
        Cout[(long)(orow + rr) * Nc + cN] = v;
      }
    }
  }
}

// ---------------- top-K (smallest distance) per row ----------------
__global__ void k_topk(const float* __restrict__ D, int* __restrict__ idxOut, int n) {
  __shared__ float row[NN];
  __shared__ float rv[256];
  __shared__ int   ri[256];
  const int tid = threadIdx.x;
  const int i   = blockIdx.x;
  const float* Dr = D + (long)i * n;
  for (int j = tid; j < n; j += 256) row[j] = Dr[j];
  __syncthreads();
  for (int t = 0; t < KNEI; ++t) {
    float bv = FLT_MAX; int bi = -1;
    for (int j = tid; j < n; j += 256) {
      float v = row[j];
      if (v < bv || (v == bv && (unsigned)j < (unsigned)bi)) { bv = v; bi = j; }
    }
    rv[tid] = bv; ri[tid] = bi;
    __syncthreads();
    for (int s = 128; s > 0; s >>= 1) {
      if (tid < s) {
        float v2 = rv[tid + s]; int i2 = ri[tid + s];
        if (v2 < rv[tid] || (v2 == rv[tid] && (unsigned)i2 < (unsigned)ri[tid])) {
          rv[tid] = v2; ri[tid] = i2;
        }
      }
      __syncthreads();
    }
    if (tid == 0) { idxOut[i * KNEI + t] = ri[0]; row[ri[0]] = FLT_MAX; }
    __syncthreads();
  }
}

// ---------------- BN column statistics ----------------
__global__ void k_colstat(const float* __restrict__ T, float* __restrict__ acc,
                          int Mrows, int C) {
  const int c  = threadIdx.x;  // blockDim.x == C
  const int r0 = blockIdx.x * 256;
  int r1 = r0 + 256; if (r1 > Mrows) r1 = Mrows;
  float s = 0.f, q = 0.f;
  for (int r = r0; r < r1; ++r) {
    float v = T[(long)r * C + c];
    s += v; q += v * v;
  }
  atomicAdd(&acc[c], s);
  atomicAdd(&acc[C + c], q);
}

__global__ void k_bn_coef(const float* __restrict__ acc, const float* __restrict__ g,
                          const float* __restrict__ be, float* __restrict__ s,
                          float* __restrict__ o, int C, float invM) {
  int c = threadIdx.x;
  if (c >= C) return;
  float mu  = acc[c] * invM;
  float var = acc[C + c] * invM - mu * mu;
  float sc  = g[c] * rsqrtf(var + 1e-5f);
  s[c] = sc;
  o[c] = be[c] - mu * sc;
}

// ---------------- max over K of relu(bn(.)) ----------------
__global__ void k_bn_relu_maxk(const float* __restrict__ T, const float* __restrict__ s,
                               const float* __restrict__ o, float* __restrict__ Hout,
                               int n, int C) {
  int t = blockIdx.x * blockDim.x + threadIdx.x;
  if (t >= n * C) return;
  int i = t / C, c = t % C;
  float sc = s[c], oc = o[c];
  float m = 0.f;  // relu outputs are >= 0
  for (int kk = 0; kk < KNEI; ++kk) {
    float v = T[(long)(i * KNEI + kk) * C + c] * sc + oc;
    m = fmaxf(m, fmaxf(v, 0.f));
  }
  Hout[(long)i * C + c] = m;
}

// ---------------- host driver ----------------
extern "C" void kernel_launch(void* const* d_in, const int* in_sizes, int n_in,
                              void* d_out, int out_size, void* d_ws, size_t ws_size,
                              hipStream_t stream) {
  (void)in_sizes; (void)n_in; (void)out_size; (void)ws_size;
  const float* features = (const float*)d_in[0];
  const int*   src = (const int*)d_in[1];
  const int*   dst = (const int*)d_in[2];
  const float* Wc1 = (const float*)d_in[3];
  const float* bc1 = (const float*)d_in[4];
  const float* Wc2 = (const float*)d_in[5];
  const float* bc2 = (const float*)d_in[6];
  const float* Wc3 = (const float*)d_in[7];
  const float* bc3 = (const float*)d_in[8];
  const float* W11 = (const float*)d_in[9];
  const float* b11 = (const float*)d_in[10];
  const float* g11 = (const float*)d_in[11];
  const float* be11 = (const float*)d_in[12];
  const float* W12 = (const float*)d_in[13];
  const float* b12 = (const float*)d_in[14];
  const float* g12 = (const float*)d_in[15];
  const float* be12 = (const float*)d_in[16];
  const float* W21 = (const float*)d_in[17];
  const float* b21 = (const float*)d_in[18];
  const float* g21 = (const float*)d_in[19];
  const float* be21 = (const float*)d_in[20];
  const float* W22 = (const float*)d_in[21];
  const float* b22 = (const float*)d_in[22];
  const float* g22 = (const float*)d_in[23];
  const float* be22 = (const float*)d_in[24];
  float* out = (float*)d_out;

  // ---- workspace layout ----
  char* w = (char*)d_ws;
  auto alloc = [&](size_t bytes) -> void* {
    void* p = (void*)w;
    w += (bytes + 255) & ~(size_t)255;
    return p;
  };
  float* D    = (float*)alloc((size_t)NN * NN * 4);          // 256 MiB; reused as t1
  float* t1   = D;                                           // alias (lifetimes disjoint)
  float* t2   = (float*)alloc((size_t)NN * KNEI * HIDC * 4); // 160 MiB
  float* h0   = (float*)alloc((size_t)NN * HIDC * 4);
  float* he   = (float*)alloc((size_t)NN * HIDC * 4);
  float* h2   = (float*)alloc((size_t)NN * HIDC * 4);
  float* agg  = (float*)alloc((size_t)NN * HIDC * 4);
  float* cntO = (float*)alloc(NN * 4);
  float* cntI = (float*)alloc(NN * 4);
  float* dO   = (float*)alloc(NN * 4);
  float* dI   = (float*)alloc(NN * 4);
  float* sq   = (float*)alloc(NN * 4);
  int*   idx  = (int*)alloc((size_t)NN * KNEI * 4);
  float* accS = (float*)alloc(2 * HIDC * 4);
  float* bnS  = (float*)alloc(HIDC * 4);
  float* bnO  = (float*)alloc(HIDC * 4);

  auto zero = [&](float* p, long n) {
    k_zero<<<dim3((unsigned)((n + 255) / 256)), dim3(256), 0, stream>>>(p, (int)n);
  };
  auto ggrid = [](int M, int Nc) { return dim3(M / 64, (Nc + 63) / 64); };

  const int MROWS = NN * KNEI;
  const float invM = 1.f / (float)MROWS;

  // ---- degree normalizations ----
  zero(cntO, NN); zero(cntI, NN);
  k_deg_accum<<<NE / 256, 256, 0, stream>>>(src, dst, cntO, cntI, NE);
  k_deg_final<<<NN / 256, 256, 0, stream>>>(cntO, dO, NN);
  k_deg_final<<<NN / 256, 256, 0, stream>>>(cntI, dI, NN);

  // ---- gconv1: features[8192,3] -> h0[8192,256], relu ----
  zero(agg, (long)NN * INDIM);
  {
    long tot = (long)NE * INDIM;
    k_gconv_scatter<<<dim3((unsigned)((tot + 255) / 256)), 256, 0, stream>>>(
        features, src, dst, dO, agg, INDIM, tot);
    k_wmma_gemm<1, true, false><<<ggrid(NN, HIDC), 128, 0, stream>>>(
        agg, Wc1, bc1, h0, NN, INDIM, HIDC, 0, dI, nullptr, nullptr, 0, nullptr, nullptr);
  }

  // ---- edge_conv 1 (256-dim features, 256-ch MLP) ----
  k_sqnorm<<<NN / 256, 256, 0, stream>>>(h0, sq, NN, HIDC);
  k_wmma_gemm<0, false, true><<<ggrid(NN, NN), 128, 0, stream>>>(
      h0, h0, sq, D, NN, HIDC, NN, HIDC, nullptr, nullptr, nullptr, 0, nullptr, nullptr);
  k_topk<<<NN, 256, 0, stream>>>(D, idx, NN);
  // layer1: t1 = [x_i, x_j - x_i] @ W11 + b11
  k_wmma_gemm<2, false, false><<<ggrid(MROWS, HIDC), 128, 0, stream>>>(
      nullptr, W11, b11, t1, MROWS, 2 * HIDC, HIDC, 0, nullptr, h0, idx, HIDC, nullptr, nullptr);
  zero(accS, 2 * HIDC);
  k_colstat<<<MROWS / 256, HIDC, 0, stream>>>(t1, accS, MROWS, HIDC);
  k_bn_coef<<<1, HIDC, 0, stream>>>(accS, g11, be11, bnS, bnO, HIDC, invM);
  // layer2: t2 = relu(bn(t1)) @ W12 + b12
  k_wmma_gemm<3, false, false><<<ggrid(MROWS, HIDC), 128, 0, stream>>>(
      t1, W12, b12, t2, MROWS, HIDC, HIDC, 0, nullptr, nullptr, nullptr, 0, bnS, bnO);
  zero(accS, 2 * HIDC);
  k_colstat<<<MROWS / 256, HIDC, 0, stream>>>(t2, accS, MROWS, HIDC);
  k_bn_coef<<<1, HIDC, 0, stream>>>(accS, g12, be12, bnS, bnO, HIDC, invM);
  k_bn_relu_maxk<<<(NN * HIDC + 255) / 256, 256, 0, stream>>>(t2, bnS, bnO, he, NN, HIDC);

  // ---- gconv2: he[8192,256] -> h2[8192,256], relu ----
  zero(agg, (long)NN * HIDC);
  {
    long tot = (long)NE * HIDC;
    k_gconv_scatter<<<dim3((unsigned)((tot + 255) / 256)), 256, 0, stream>>>(
        he, src, dst, dO, agg, HIDC, tot);
    k_wmma_gemm<1, true, false><<<ggrid(NN, HIDC), 128, 0, stream>>>(
        agg, Wc2, bc2, h2, NN, HIDC, HIDC, 0, dI, nullptr, nullptr, 0, nullptr, nullptr);
  }

  // ---- edge_conv 2 (256-dim features, 64-ch MLP) ----
  k_sqnorm<<<NN / 256, 256, 0, stream>>>(h2, sq, NN, HIDC);
  k_wmma_gemm<0, false, true><<<ggrid(NN, NN), 128, 0, stream>>>(
      h2, h2, sq, D, NN, HIDC, NN, HIDC, nullptr, nullptr, nullptr, 0, nullptr, nullptr);
  k_topk<<<NN, 256, 0, stream>>>(D, idx, NN);
  k_wmma_gemm<2, false, false><<<ggrid(MROWS, 64), 128, 0, stream>>>(
      nullptr, W21, b21, t1, MROWS, 2 * HIDC, 64, 0, nullptr, h2, idx, HIDC, nullptr, nullptr);
  zero(accS, 2 * 64);
  k_colstat<<<MROWS / 256, 64, 0, stream>>>(t1, accS, MROWS, 64);
  k_bn_coef<<<1, 64, 0, stream>>>(accS, g21, be21, bnS, bnO, 64, invM);
  k_wmma_gemm<3, false, false><<<ggrid(MROWS, 64), 128, 0, stream>>>(
      t1, W22, b22, t2, MROWS, 64, 64, 0, nullptr, nullptr, nullptr, 0, bnS, bnO);
  zero(accS, 2 * 64);
  k_colstat<<<MROWS / 256, 64, 0, stream>>>(t2, accS, MROWS, 64);
  k_bn_coef<<<1, 64, 0, stream>>>(accS, g22, be22, bnS, bnO, 64, invM);
  k_bn_relu_maxk<<<(NN * 64 + 255) / 256, 256, 0, stream>>>(t2, bnS, bnO, he, NN, 64);

  // ---- gconv3: he[8192,64] -> out[8192,32], no activation ----
  zero(agg, (long)NN * 64);
  {
    long tot = (long)NE * 64;
    k_gconv_scatter<<<dim3((unsigned)((tot + 255) / 256)), 256, 0, stream>>>(
        he, src, dst, dO, agg, 64, tot);
    k_wmma_gemm<1, false, false><<<ggrid(NN, NCLS), 128, 0, stream>>>(
        agg, Wc3, bc3, out, NN, 64, NCLS, 0, dI, nullptr, nullptr, 0, nullptr, nullptr);
  }
}